// ProbSparseAttention_1340029796602
// MI455X (gfx1250) — compile-verified
//
#include <hip/hip_runtime.h>

typedef __attribute__((ext_vector_type(2))) float v2f;
typedef __attribute__((ext_vector_type(8))) float v8f;
typedef __attribute__((ext_vector_type(4))) unsigned int u32x4;
typedef __attribute__((ext_vector_type(8))) int i32x8;
typedef __attribute__((ext_vector_type(4))) int i32x4;

#define B_ 4
#define L_ 2048
#define H_ 8
#define D_ 64
#define SAMPK 40
#define NSEL 40
#define NCHUNK 16
#define CHUNK (L_ / NCHUNK)
#define ROWSTRIDE (H_ * D_)   // 512 floats between consecutive L rows
#define SCALE 0.125f          // 1/sqrt(64)
#define LDSROW 66             // 64 + 2 pad floats per row (TDM pad; keeps 8B alignment)
#define PROW 18               // pbuf row stride (8B-aligned pairs, conflict-free)

// ---------------------------------------------------------------------------
// TDM: load one 16-row x 64-float f32 tile (row stride 512 floats) into LDS,
// inserting 2 pad DWORDs per 64 DWORDs (-> 66-float LDS row stride).
// D# per CDNA5 ISA ch.8; groups 2/3 zero (2-D tensor).
// ---------------------------------------------------------------------------
__device__ __forceinline__ void tdm_load_tile(unsigned lds_addr, const float* gptr) {
    unsigned long long ga = (unsigned long long)(uintptr_t)gptr;
    u32x4 g0;
    g0[0] = 1u;                                                   // count=1 (valid), user mode
    g0[1] = lds_addr;                                             // LDS byte address
    g0[2] = (unsigned)(ga & 0xFFFFFFFFu);                         // global_addr[31:0]
    g0[3] = (unsigned)((ga >> 32) & 0x01FFFFFFu) | (2u << 30);    // addr[56:32] | type=2
    i32x8 g1;
    // data_size=4B | pad_enable | pad_interval=5 (64 DW) | pad_amount=1 (2 DW)
    g1[0] = (int)((2u << 16) | (1u << 20) | (5u << 22) | (1u << 25));
    g1[1] = (int)(64u << 16);                                     // tensor_dim0 = 64
    g1[2] = (int)(2048u << 16);                                   // tensor_dim1 = 2048
    g1[3] = (int)(64u << 16);                                     // tile_dim0 = 64
    g1[4] = 16;                                                   // tile_dim1 = 16 rows
    g1[5] = 512;                                                  // tensor_dim0_stride = 512
    g1[6] = 0;
    g1[7] = 0;
    i32x4 z4 = { 0, 0, 0, 0 };
    i32x8 z8 = { 0, 0, 0, 0, 0, 0, 0, 0 };
    __builtin_amdgcn_tensor_load_to_lds(g0, g1, z4, z4, z8, 0);
}

// ---------------------------------------------------------------------------
// Kernel 1: sparsity measure M[b,h,l] = max_s(q.k_s) - sum_s(q.k_s)/L
// ---------------------------------------------------------------------------
__global__ void __launch_bounds__(256, 1)
msum_kernel(const float* __restrict__ Q, const float* __restrict__ K,
            const long long* __restrict__ idxs, float* __restrict__ M) {
    int t = blockIdx.x * blockDim.x + threadIdx.x;   // t = b*L*H + l*H + h
    if (t >= B_ * L_ * H_) return;
    int b = t / (L_ * H_);
    int l = (t / H_) % L_;
    int h = t % H_;

    const float4* qp = (const float4*)(Q + (size_t)t * D_);
    float4 qr[16];
#pragma unroll
    for (int j = 0; j < 16; ++j) qr[j] = qp[j];

    float maxv = -__builtin_inff();
    float sumv = 0.f;
    for (int s = 0; s < SAMPK; ++s) {
        long long kidx = idxs[(size_t)l * SAMPK + s];
        const float4* kp = (const float4*)(K + (((size_t)b * L_ + (size_t)kidx) * H_ + h) * D_);
        float dot = 0.f;
#pragma unroll
        for (int j = 0; j < 16; ++j) {
            float4 kv = kp[j];
            dot += qr[j].x * kv.x + qr[j].y * kv.y + qr[j].z * kv.z + qr[j].w * kv.w;
        }
        maxv = fmaxf(maxv, dot);
        sumv += dot;
    }
    M[((size_t)b * H_ + h) * L_ + l] = maxv - sumv * (1.0f / (float)L_);
}

// ---------------------------------------------------------------------------
// Kernel 2: top-40 per (b,h): iterative argmax over LDS-resident M row
// ---------------------------------------------------------------------------
__global__ void __launch_bounds__(256, 1)
topk_kernel(const float* __restrict__ M, int* __restrict__ qidx) {
    int bh = blockIdx.x;
    __shared__ float sm[L_];
    __shared__ float redv[256];
    __shared__ int   redi[256];
    const float* Mp = M + (size_t)bh * L_;
    for (int i = threadIdx.x; i < L_; i += blockDim.x) sm[i] = Mp[i];
    __syncthreads();
    for (int it = 0; it < NSEL; ++it) {
        float bv = -__builtin_inff(); int bi = 0;
        for (int i = threadIdx.x; i < L_; i += blockDim.x) {
            float v = sm[i];
            if (v > bv) { bv = v; bi = i; }
        }
        redv[threadIdx.x] = bv; redi[threadIdx.x] = bi;
        __syncthreads();
        for (int off = 128; off > 0; off >>= 1) {
            if ((int)threadIdx.x < off) {
                float ov = redv[threadIdx.x + off]; int oi = redi[threadIdx.x + off];
                if (ov > redv[threadIdx.x] || (ov == redv[threadIdx.x] && oi < redi[threadIdx.x])) {
                    redv[threadIdx.x] = ov; redi[threadIdx.x] = oi;
                }
            }
            __syncthreads();
        }
        if (threadIdx.x == 0) { qidx[bh * NSEL + it] = redi[0]; sm[redi[0]] = -__builtin_inff(); }
        __syncthreads();
    }
}

// ---------------------------------------------------------------------------
// Kernel 3a/3b: chunked inclusive prefix-sum of V along L (default output)
// ---------------------------------------------------------------------------
__global__ void __launch_bounds__(64, 1)
cumsum_part(const float* __restrict__ V, float* __restrict__ out,
            float* __restrict__ totals) {
    int blk = blockIdx.x;
    int bh = blk / NCHUNK, ck = blk % NCHUNK;
    int b = bh / H_, h = bh % H_;
    int d = threadIdx.x;
    float acc = 0.f;
    int l0 = ck * CHUNK;
    for (int l = l0; l < l0 + CHUNK; ++l) {
        size_t o = (((size_t)b * L_ + l) * H_ + h) * D_ + d;
        acc += V[o];
        out[o] = acc;
    }
    totals[((size_t)bh * NCHUNK + ck) * D_ + d] = acc;
}

__global__ void __launch_bounds__(64, 1)
cumsum_fix(float* __restrict__ out, const float* __restrict__ totals) {
    int blk = blockIdx.x;
    int bh = blk / NCHUNK, ck = blk % NCHUNK;
    if (ck == 0) return;
    int b = bh / H_, h = bh % H_;
    int d = threadIdx.x;
    float off = 0.f;
    for (int c = 0; c < ck; ++c) off += totals[((size_t)bh * NCHUNK + c) * D_ + d];
    int l0 = ck * CHUNK;
    for (int l = l0; l < l0 + CHUNK; ++l) {
        size_t o = (((size_t)b * L_ + l) * H_ + h) * D_ + d;
        out[o] += off;
    }
}

// ---------------------------------------------------------------------------
// Kernel 4: streaming causal attention for the 40 selected queries per (b,h).
// V_WMMA_F32_16X16X4_F32 for QK^T and P*V; K/V tiles double-buffered in LDS
// via the Tensor Data Mover (TENSORcnt), issued by wave 0 for the block.
// Block = 96 threads (3 waves); wave w owns padded rows [16w, 16w+16).
// ---------------------------------------------------------------------------
__global__ void __launch_bounds__(96, 1)
attn_kernel(const float* __restrict__ Q, const float* __restrict__ K,
            const float* __restrict__ V, const int* __restrict__ qidx,
            float* __restrict__ out) {
    int bh = blockIdx.x;
    int b = bh / H_, h = bh % H_;
    int lane = threadIdx.x & 31;
    int wave = threadIdx.x >> 5;        // 0..2
    int half = lane >> 4;               // 0 | 1
    int l15  = lane & 15;
    int koff = half * 2;                // K-subcolumn offset per f32 WMMA A/B layout

    __shared__ float klds[2][16 * LDSROW];   // double-buffered K tile (padded rows)
    __shared__ float vlds[2][16 * LDSROW];   // double-buffered V tile
    __shared__ float pbuf[3][16 * PROW];     // per-wave P tile (C layout -> A layout)

    // selected-query index per C-layout row (VGPR g <-> row g + 8*half)
    int qrow[8];
#pragma unroll
    for (int g = 0; g < 8; ++g) {
        int r = wave * 16 + g + half * 8;
        qrow[g] = (r < NSEL) ? qidx[bh * NSEL + r] : 0;
    }
    // A-layout row for this lane (row m = lane&15)
    int rA = wave * 16 + l15;
    int qA = (rA < NSEL) ? qidx[bh * NSEL + rA] : 0;

    // Preload the 16 (K=4) A-operand pairs of this lane's Q row
    v2f qa[16];
    {
        const float* qp = Q + (((size_t)b * L_ + qA) * H_ + h) * D_;
#pragma unroll
        for (int k = 0; k < 16; ++k) {
            qa[k].x = qp[4 * k + koff];
            qa[k].y = qp[4 * k + koff + 1];
        }
    }

    const float* Kbase = K + ((size_t)b * L_ * H_ + h) * D_;
    const float* Vbase = V + ((size_t)b * L_ * H_ + h) * D_;

    v8f o0 = {}, o1 = {}, o2 = {}, o3 = {};
    float mrow[8], lrow[8];
#pragma unroll
    for (int g = 0; g < 8; ++g) { mrow[g] = -__builtin_inff(); lrow[g] = 0.f; }

    const int NT = L_ / 16;

    // prologue: stage tile 0 into buffer 0
    if (wave == 0) {
        tdm_load_tile((unsigned)(uintptr_t)&klds[0][0], Kbase);
        tdm_load_tile((unsigned)(uintptr_t)&vlds[0][0], Vbase);
    }

    for (int t = 0; t < NT; ++t) {
        int p = t & 1;
        if (wave == 0) {
            if (t + 1 < NT) {
                const float* kg = Kbase + (size_t)(t + 1) * 16 * ROWSTRIDE;
                const float* vg = Vbase + (size_t)(t + 1) * 16 * ROWSTRIDE;
                tdm_load_tile((unsigned)(uintptr_t)&klds[p ^ 1][0], kg);
                tdm_load_tile((unsigned)(uintptr_t)&vlds[p ^ 1][0], vg);
                __builtin_amdgcn_s_wait_tensorcnt(2);   // tile t's pair retired (in-order)
            } else {
                __builtin_amdgcn_s_wait_tensorcnt(0);
            }
        }
        __syncthreads();                // tile t visible to all waves

        int key_base = t * 16;

        // ---- S = Q * K^T : batch all 16 B-operand b64 loads, then 16 WMMAs ----
        v2f kb[16];
        const float* kp = &klds[p][l15 * LDSROW + koff];   // key = key_base + l15
#pragma unroll
        for (int k = 0; k < 16; ++k) kb[k] = *(const v2f*)(kp + 4 * k);
        __builtin_amdgcn_sched_barrier(0);   // keep loads batched ahead of the WMMA chain
        v8f s = {};
#pragma unroll
        for (int k = 0; k < 16; ++k) {
            s = __builtin_amdgcn_wmma_f32_16x16x4_f32(false, qa[k], false, kb[k],
                                                      (short)0, s, false, false);
        }

        // ---- scale, causal mask, online softmax (rows live in lane halves) ----
        int keyB = key_base + l15;
        float pv[8], sc[8];
#pragma unroll
        for (int g = 0; g < 8; ++g) {
            float v = s[g] * SCALE;
            if (keyB > qrow[g]) v = -__builtin_inff();
            float mx = v;
            mx = fmaxf(mx, __shfl_xor(mx, 1, 16));
            mx = fmaxf(mx, __shfl_xor(mx, 2, 16));
            mx = fmaxf(mx, __shfl_xor(mx, 4, 16));
            mx = fmaxf(mx, __shfl_xor(mx, 8, 16));
            float nm = fmaxf(mrow[g], mx);
            float so = __expf(mrow[g] - nm);
            float pe = __expf(v - nm);
            float ps = pe;
            ps += __shfl_xor(ps, 1, 16);
            ps += __shfl_xor(ps, 2, 16);
            ps += __shfl_xor(ps, 4, 16);
            ps += __shfl_xor(ps, 8, 16);
            lrow[g] = lrow[g] * so + ps;
            mrow[g] = nm;
            pv[g] = pe;
            sc[g] = so;
        }
#pragma unroll
        for (int g = 0; g < 8; ++g) {
            o0[g] *= sc[g]; o1[g] *= sc[g]; o2[g] *= sc[g]; o3[g] *= sc[g];
        }

        // ---- stage P (C layout) -> LDS -> reread in A layout ----
#pragma unroll
        for (int g = 0; g < 8; ++g) pbuf[wave][(g + half * 8) * PROW + l15] = pv[g];
        __syncthreads();                // pbuf RAW: cross-lane visibility

        // ---- O += P * V : batch operands, then 4 d-blocks x 4 K-steps WMMA ----
        v2f pa[4];
#pragma unroll
        for (int kk = 0; kk < 4; ++kk)
            pa[kk] = *(const v2f*)&pbuf[wave][l15 * PROW + 4 * kk + koff];
        v2f vb[4][4];
#pragma unroll
        for (int nb = 0; nb < 4; ++nb) {
#pragma unroll
            for (int kk = 0; kk < 4; ++kk) {
                int kr = 4 * kk + koff;     // key row for B VGPR0; +1 for VGPR1
                vb[nb][kk].x = vlds[p][kr * LDSROW + nb * 16 + l15];
                vb[nb][kk].y = vlds[p][(kr + 1) * LDSROW + nb * 16 + l15];
            }
        }
        __builtin_amdgcn_sched_barrier(0);   // loads batched ahead of WMMA chain
        v8f oacc[4] = { o0, o1, o2, o3 };
#pragma unroll
        for (int nb = 0; nb < 4; ++nb) {
#pragma unroll
            for (int kk = 0; kk < 4; ++kk) {
                oacc[nb] = __builtin_amdgcn_wmma_f32_16x16x4_f32(false, pa[kk], false, vb[nb][kk],
                                                                 (short)0, oacc[nb], false, false);
            }
        }
        o0 = oacc[0]; o1 = oacc[1]; o2 = oacc[2]; o3 = oacc[3];

        __syncthreads();                // all reads of buffer p / pbuf done before reuse
    }

    // ---- normalize and scatter the 40 valid rows ----
#pragma unroll
    for (int g = 0; g < 8; ++g) {
        int r = wave * 16 + g + half * 8;
        if (r < NSEL) {
            float inv = 1.0f / lrow[g];
            size_t base = (((size_t)b * L_ + qrow[g]) * H_ + h) * D_;
            out[base +  0 + l15] = o0[g] * inv;
            out[base + 16 + l15] = o1[g] * inv;
            out[base + 32 + l15] = o2[g] * inv;
            out[base + 48 + l15] = o3[g] * inv;
        }
    }
}

// ---------------------------------------------------------------------------
extern "C" void kernel_launch(void* const* d_in, const int* in_sizes, int n_in,
                              void* d_out, int out_size, void* d_ws, size_t ws_size,
                              hipStream_t stream) {
    const float*      Q    = (const float*)d_in[0];
    const float*      K    = (const float*)d_in[1];
    const float*      V    = (const float*)d_in[2];
    const long long*  idxs = (const long long*)d_in[3];
    float*            out  = (float*)d_out;

    char* ws = (char*)d_ws;
    float* Mbuf   = (float*)(ws);                         // B*H*L floats  (256 KB)
    float* totals = (float*)(ws + 262144);                // B*H*NCHUNK*D  (128 KB)
    int*   qsel   = (int*)  (ws + 262144 + 131072);       // B*H*NSEL ints (5 KB)

    // 1. sparsity measure
    msum_kernel<<<(B_ * L_ * H_) / 256, 256, 0, stream>>>(Q, K, idxs, Mbuf);
    // 2. top-40 queries per (b,h)
    topk_kernel<<<B_ * H_, 256, 0, stream>>>(Mbuf, qsel);
    // 3. default output: cumsum(V) along L (chunked 2-phase scan)
    cumsum_part<<<B_ * H_ * NCHUNK, D_, 0, stream>>>(V, out, totals);
    cumsum_fix <<<B_ * H_ * NCHUNK, D_, 0, stream>>>(out, totals);
    // 4. WMMA + TDM causal attention for selected queries, scattered over cumsum
    attn_kernel<<<B_ * H_, 96, 0, stream>>>(Q, K, V, qsel, out);
}